// SparseNet_79053168050211
// MI455X (gfx1250) — compile-verified
//
#include <hip/hip_runtime.h>

typedef __attribute__((ext_vector_type(2))) float v2f;
typedef __attribute__((ext_vector_type(8))) float v8f;

// ---------------------------------------------------------------------------
// Generic LDS-tiled transpose: src[R,C] row-major -> dst[C,R].
// Coalesced on both sides; 16x16 tiles, padded LDS (no bank conflicts).
// ---------------------------------------------------------------------------
__global__ void k_transpose(const float* __restrict__ src,
                            float* __restrict__ dst, int R, int C) {
    __shared__ float tile[16][17];
    const int c0 = blockIdx.x * 16;
    const int r0 = blockIdx.y * 16;
    const int tx = threadIdx.x, ty = threadIdx.y;
    int c = c0 + tx, r = r0 + ty;
    if (r < R && c < C) tile[ty][tx] = src[(size_t)r * C + c];
    __syncthreads();
    c = c0 + ty; r = r0 + tx;
    if (c < C && r < R) dst[(size_t)c * R + r] = tile[tx][ty];
}

// ---------------------------------------------------------------------------
// gene[G,B] = broadcast(gene_bias[G]) across batch. One block per gene row.
// ---------------------------------------------------------------------------
__global__ void k_init_gene(float* __restrict__ gene,
                            const float* __restrict__ bias, int B) {
    const int g = blockIdx.x;
    const float v = bias[g];
    for (int b = threadIdx.x; b < B; b += blockDim.x)
        gene[(size_t)g * B + b] = v;
}

// ---------------------------------------------------------------------------
// Edge scatter, coalesced path: xT[S,B] so the gather x[:,si] is eight 128B
// contiguous wave reads; gene[G,B] so the 32 lane atomics share one L2 line.
// One wave per edge; lanes sweep the batch dimension.
// ---------------------------------------------------------------------------
__global__ void k_edge_scatter_xt(const float* __restrict__ xT,
                                  const int* __restrict__ snp,
                                  const int* __restrict__ gidx,
                                  const float* __restrict__ ew,
                                  float* __restrict__ gene,
                                  int B, int E) {
    const int wave = threadIdx.x >> 5;          // 8 waves / block (wave32)
    const int lane = threadIdx.x & 31;
    const int e = blockIdx.x * 8 + wave;
    if (e >= E) return;
    const float w = ew[e];
    const float* __restrict__ xr = xT   + (size_t)snp[e]  * B;  // contiguous row
    float* __restrict__       gr = gene + (size_t)gidx[e] * B;
    for (int b = lane; b < B; b += 32)
        atomicAdd(&gr[b], xr[b] * w);
}

// Fallback (workspace too small to hold xT): lane-strided gather, x L2-resident.
__global__ void k_edge_scatter_direct(const float* __restrict__ x,
                                      const int* __restrict__ snp,
                                      const int* __restrict__ gidx,
                                      const float* __restrict__ ew,
                                      float* __restrict__ gene,
                                      int B, int S, int E) {
    const int wave = threadIdx.x >> 5;
    const int lane = threadIdx.x & 31;
    const int e = blockIdx.x * 8 + wave;
    if (e >= E) return;
    const float w = ew[e];
    const float* __restrict__ xc = x    + snp[e];
    float* __restrict__       gr = gene + (size_t)gidx[e] * B;
    for (int b = lane; b < B; b += 32)
        atomicAdd(&gr[b], xc[(size_t)b * S] * w);
}

// ---------------------------------------------------------------------------
// h = gene @ W1^T + b1 in full fp32 via V_WMMA_F32_16X16X4_F32.
//   A (16x4, M=batch,K=g) from gene[G,B]; B (4x16) from W1T[G,H].
// All four fragment loads use sgpr-base + 32-bit voffset (scale_offset):
// the second element of each fragment gets its own base pointer (gene+B,
// W1T+H) so it can reuse the SAME index register — no 64-bit address math
// in the hot loop, just two v_add_nc_u32 per k-step. Unroll x8: 32 loads
// in flight against 8 back-to-back WMMAs.
// grid.x = B/16 strips; blockDim = (H/16) full waves (EXEC all-ones).
// ---------------------------------------------------------------------------
__global__ void __launch_bounds__(512)
k_gemm_wmma(const float* __restrict__ gene,
            const float* __restrict__ W1T,
            const float* __restrict__ b1,
            float* __restrict__ h,
            int B, int G, int H) {
    const int lane = threadIdx.x & 31;
    const int wv   = threadIdx.x >> 5;
    const int n0   = wv * 16;
    const int b0   = blockIdx.x * 16;
    const int half = lane >> 4;            // 0: lanes 0-15, 1: lanes 16-31
    const int l15  = lane & 15;
    const int m    = b0 + l15;             // A row (batch)
    const int n    = n0 + l15;             // B col (hidden unit)

    // Second-element bases: same 32-bit index, different sgpr base.
    const float* __restrict__ geneB = gene + B;
    const float* __restrict__ W1TH  = W1T + H;

    int ia = (2 * half) * B + m;           // gene[(k+2h)*B + m]
    int ib = (2 * half) * H + n;           // W1T [(k+2h)*H + n]
    const int dA = 4 * B, dB = 4 * H;

    v8f c = {0.f, 0.f, 0.f, 0.f, 0.f, 0.f, 0.f, 0.f};

#pragma unroll 8
    for (int k0 = 0; k0 < G; k0 += 4) {
        v2f a, bf;
        a.x  = gene [ia];
        a.y  = geneB[ia];
        bf.x = W1T  [ib];
        bf.y = W1TH [ib];
        c = __builtin_amdgcn_wmma_f32_16x16x4_f32(
                false, a, false, bf, (short)0, c, false, false);
        ia += dA; ib += dB;
    }

    const float bias = b1[n];
    int io = (b0 + 8 * half) * H + n;      // C/D layout: VGPR r -> M = r + 8*half
#pragma unroll
    for (int r = 0; r < 8; ++r) {
        h[io] = c[r] + bias;
        io += H;
    }
}

// ---------------------------------------------------------------------------
// out[b] = h[b,:] . W2 + b2   (tiny)
// ---------------------------------------------------------------------------
__global__ void k_out(const float* __restrict__ h,
                      const float* __restrict__ W2,
                      const float* __restrict__ b2,
                      float* __restrict__ out, int B, int H) {
    const int b = blockIdx.x * blockDim.x + threadIdx.x;
    if (b >= B) return;
    const float* hr = h + (size_t)b * H;
    float s = 0.f;
    for (int j = 0; j < H; ++j) s = fmaf(hr[j], W2[j], s);
    out[b] = s + b2[0];
}

// ---------------------------------------------------------------------------
// Launch.  Inputs: x[B*S], snp_idx[E], gene_idx[E], edge_w[E], gene_bias[G],
//                  W1[H*G], b1[H], W2[H], b2[1];  out: [B]
// ws: gene[G*B] | W1T[G*H] | h[B*H] | (optional) xT[S*B]
// ---------------------------------------------------------------------------
extern "C" void kernel_launch(void* const* d_in, const int* in_sizes, int n_in,
                              void* d_out, int out_size, void* d_ws, size_t ws_size,
                              hipStream_t stream) {
    const float* x    = (const float*)d_in[0];
    const int*   snp  = (const int*)  d_in[1];
    const int*   gidx = (const int*)  d_in[2];
    const float* ew   = (const float*)d_in[3];
    const float* gb   = (const float*)d_in[4];
    const float* W1   = (const float*)d_in[5];
    const float* b1   = (const float*)d_in[6];
    const float* W2   = (const float*)d_in[7];
    const float* b2   = (const float*)d_in[8];
    float*       out  = (float*)d_out;

    const int B = out_size;
    const int S = in_sizes[0] / B;
    const int E = in_sizes[1];
    const int G = in_sizes[4];
    const int H = in_sizes[6];

    float* gene = (float*)d_ws;                    // [G,B]
    float* W1T  = gene + (size_t)G * B;            // [G,H]
    float* h    = W1T  + (size_t)G * H;            // [B,H]
    float* xT   = h    + (size_t)B * H;            // [S,B] (optional)

    const size_t base_floats = (size_t)G * B + (size_t)G * H + (size_t)B * H;
    const size_t xt_floats   = (size_t)S * B;
    const bool use_xt = ws_size >= (base_floats + xt_floats) * sizeof(float);

    // 1) W1[H,G] -> W1T[G,H]
    k_transpose<<<dim3((G + 15) / 16, (H + 15) / 16), dim3(16, 16), 0, stream>>>(
        W1, W1T, H, G);

    // 1b) x[B,S] -> xT[S,B] (if workspace allows; makes edge gathers coalesced)
    if (use_xt)
        k_transpose<<<dim3((S + 15) / 16, (B + 15) / 16), dim3(16, 16), 0, stream>>>(
            x, xT, B, S);

    // 2) gene = broadcast(gene_bias)
    k_init_gene<<<G, 256, 0, stream>>>(gene, gb, B);

    // 3) scatter edges (coalesced wave-wide atomics, one wave per edge)
    if (use_xt)
        k_edge_scatter_xt<<<(E + 7) / 8, 256, 0, stream>>>(xT, snp, gidx, ew,
                                                           gene, B, E);
    else
        k_edge_scatter_direct<<<(E + 7) / 8, 256, 0, stream>>>(x, snp, gidx, ew,
                                                               gene, B, S, E);

    // 4) h = gene @ W1^T + b1  (fp32 WMMA)
    k_gemm_wmma<<<B / 16, (H / 16) * 32, 0, stream>>>(gene, W1T, b1, h, B, G, H);

    // 5) out = h @ W2^T + b2
    k_out<<<(B + 255) / 256, 256, 0, stream>>>(h, W2, b2, out, B, H);
}